// SSMKernel_56770877718744
// MI455X (gfx1250) — compile-verified
//
#include <hip/hip_runtime.h>

// kernel[l, i] = exp(l * A[i][i]) * B[i] * C[i] + D
// L = 262144, n = 256 -> output (L, n) fp32 = 256 MB. Arithmetic intensity
// ~0.08 flop/byte: pure store-bandwidth bound, floor ~= 268MB / 23.3TB/s ~= 11.5us.
// No contraction dimension exists, so WMMA is inapplicable; the CDNA5 path that
// matters is v_exp_f32 (TRANS, co-executes with VALU) + nontemporal
// global_store_b128 (256 MB output > 192 MB L2 -> stream past cache).

typedef float v4f __attribute__((ext_vector_type(4)));

#define SSM_LOG2E 1.4426950408889634f

__global__ void ssm_precompute_kernel(const float* __restrict__ A,
                                      const float* __restrict__ B,
                                      const float* __restrict__ C,
                                      float* __restrict__ ws, int n) {
    int i = blockIdx.x * blockDim.x + threadIdx.x;
    if (i < n) {
        // exp(l * a) == exp2(l * (a * log2e)) -> one v_exp_f32 per element later
        ws[i]     = A[(size_t)i * n + i] * SSM_LOG2E;  // log2 of per-step decay
        ws[n + i] = B[i] * C[i];
    }
}

template <int N>
__global__ __launch_bounds__(256) void ssm_stream_kernel(
    const float* __restrict__ ws,   // [0..N): a2 = A_ii*log2e, [N..2N): bc = B*C
    const float* __restrict__ Dp,   // scalar D (uniform s_load)
    float* __restrict__ out,        // (L, N) row-major
    int L) {
    constexpr int COLG = N / 4;              // float4 column groups per row (64)
    constexpr int RPP  = 256 / COLG;         // rows per pass per block (4)
    constexpr int UNROLL = 8;
    constexpr int ROWS_PER_BLOCK = RPP * UNROLL;  // 32

    const int tid = threadIdx.x;
    const int c4  = tid & (COLG - 1);        // column group: i = 4*c4 .. 4*c4+3
    const int r0  = tid / COLG;              // 0..RPP-1

    // gfx1250 global_prefetch_b8 on the 2 KB coefficient table (L2-resident).
    __builtin_prefetch(ws + tid * 2, 0, 3);

    // Two B128 loads per thread, amortized over 8 B128 stores.
    const v4f a2 = ((const v4f*)ws)[c4];
    const v4f bc = ((const v4f*)(ws + N))[c4];
    const float d = Dp[0];                   // uniform -> scalar load

    const int lBase = blockIdx.x * ROWS_PER_BLOCK + r0;
    v4f* __restrict__ p = (v4f*)out + (size_t)lBase * COLG + c4;
    const float lf0 = (float)lBase;          // exact: l < 2^24

    if (lBase + (UNROLL - 1) * RPP < L) {
        // Fast path (always taken when L % 32 == 0): no exec-mask churn,
        // addresses fold into the 24-bit instruction offset (r * 4096 B).
#pragma unroll
        for (int r = 0; r < UNROLL; ++r) {
            const float lf = lf0 + (float)(r * RPP);
            v4f v;
            v.x = __builtin_amdgcn_exp2f(lf * a2.x) * bc.x + d;
            v.y = __builtin_amdgcn_exp2f(lf * a2.y) * bc.y + d;
            v.z = __builtin_amdgcn_exp2f(lf * a2.z) * bc.z + d;
            v.w = __builtin_amdgcn_exp2f(lf * a2.w) * bc.w + d;
            __builtin_nontemporal_store(v, p + (size_t)r * RPP * COLG);
        }
    } else {
        // Generic guarded tail (unused for L % 32 == 0, kept for correctness).
#pragma unroll
        for (int r = 0; r < UNROLL; ++r) {
            const int l = lBase + r * RPP;
            if (l < L) {
                const float lf = (float)l;
                v4f v;
                v.x = __builtin_amdgcn_exp2f(lf * a2.x) * bc.x + d;
                v.y = __builtin_amdgcn_exp2f(lf * a2.y) * bc.y + d;
                v.z = __builtin_amdgcn_exp2f(lf * a2.z) * bc.z + d;
                v.w = __builtin_amdgcn_exp2f(lf * a2.w) * bc.w + d;
                __builtin_nontemporal_store(v, p + (size_t)r * RPP * COLG);
            }
        }
    }
}

extern "C" void kernel_launch(void* const* d_in, const int* in_sizes, int n_in,
                              void* d_out, int out_size, void* d_ws, size_t ws_size,
                              hipStream_t stream) {
    // Inputs (setup_inputs order): L (int scalar), A (n,n), B (n,1), C (1,n), D (1,)
    const float* A = (const float*)d_in[1];
    const float* B = (const float*)d_in[2];
    const float* C = (const float*)d_in[3];
    const float* D = (const float*)d_in[4];
    float* out = (float*)d_out;
    float* ws  = (float*)d_ws;

    const int n = in_sizes[2];        // B has n elements (n == 256)
    const int L = out_size / n;       // host-side, graph-capture safe

    ssm_precompute_kernel<<<(n + 255) / 256, 256, 0, stream>>>(A, B, C, ws, n);

    constexpr int ROWS_PER_BLOCK = 32;
    const int nBlocks = (L + ROWS_PER_BLOCK - 1) / ROWS_PER_BLOCK;
    ssm_stream_kernel<256><<<nBlocks, 256, 0, stream>>>(ws, D, out, L);
}